// Net_1460288881142
// MI455X (gfx1250) — compile-verified
//
#include <hip/hip_runtime.h>

typedef __attribute__((ext_vector_type(2))) float v2f;
typedef __attribute__((ext_vector_type(8))) float v8f;

#if __has_builtin(__builtin_amdgcn_global_load_async_to_lds_b32)
#define USE_ASYNC_LDS 1
#else
#define USE_ASYNC_LDS 0
#endif

namespace {
constexpr float kBeta  = 0.9f;
constexpr float kTh    = 1.0f;
constexpr int   kSteps = 8192;
constexpr int   kB     = 128;
constexpr int   kLvl   = 4;
constexpr int   kCh    = 23;
constexpr int   kDin   = 92;   // kLvl * kCh
constexpr int   kH     = 8;
constexpr int   kDout  = 2;
constexpr int   kKB    = 23;   // 92/4 k-blocks for wmma 16x16x4
constexpr int   kTileM = 16;
constexpr int   kElems = kTileM * kDin;  // 1472 floats per step tile
constexpr int   kT     = 8;              // steps per barrier group
constexpr int   kG     = kSteps / kT;    // 1024 groups
constexpr int   kSpkStride = 17;         // padded stride for spk transpose tile
}

#if USE_ASYNC_LDS
__device__ static inline void async_ld_f32(const float* g, float* l) {
    __builtin_amdgcn_global_load_async_to_lds_b32(
        (__attribute__((address_space(1))) int*)const_cast<float*>(g),
        (__attribute__((address_space(3))) int*)l,
        0, 0);
}
__device__ static inline void wait_async0() {
#if __has_builtin(__builtin_amdgcn_s_wait_asynccnt)
    __builtin_amdgcn_s_wait_asynccnt(0);
#else
    asm volatile("s_wait_asynccnt 0" ::: "memory");
#endif
}
#endif

__global__ __launch_bounds__(128, 1)
void snn_scan_kernel(const float* __restrict__ x,
                     const float* __restrict__ W1,
                     const float* __restrict__ b1,
                     const float* __restrict__ W2,
                     const float* __restrict__ b2,
                     float* __restrict__ out)
{
    // double-buffered group of kT step tiles + spk1 transpose scratch
    __shared__ __align__(16) float tile[2][kT][kElems];
    __shared__ __align__(16) float spkT[16 * kSpkStride];

    const int tid = threadIdx.x;
    const int b0  = blockIdx.x * kTileM;

    if (tid < 32) {
        // ---------------- compute wave (wave 0) ----------------
        const int n  = tid & 15;   // N column / A row M
        const int hi = tid >> 4;   // K-pair select: 0 -> K=0,1 ; 1 -> K=2,3

        // Layer-1 B fragments: W1^T (92x16), cols 8..15 zero padded
        v2f bfrag[kKB];
#pragma unroll
        for (int kb = 0; kb < kKB; ++kb) {
            const int d0 = 4 * kb + 2 * hi;
            v2f t;
            t.x = (n < kH) ? W1[n * kDin + d0]     : 0.0f;
            t.y = (n < kH) ? W1[n * kDin + d0 + 1] : 0.0f;
            bfrag[kb] = t;
        }
        const float b1n = (n < kH) ? b1[n] : 0.0f;

        // Layer-2 A fragments: W2 (2x8) zero-padded to 16x8, K-blocks {0..3},{4..7}
        v2f w2a[2];
#pragma unroll
        for (int kb = 0; kb < 2; ++kb) {
            const int k0 = 4 * kb + 2 * hi;
            v2f t;
            t.x = (n < kDout) ? W2[n * kH + k0]     : 0.0f;
            t.y = (n < kDout) ? W2[n * kH + k0 + 1] : 0.0f;
            w2a[kb] = t;
        }
        const float b20 = b2[0];
        const float b21 = b2[1];

        v8f mem1;
#pragma unroll
        for (int j = 0; j < 8; ++j) mem1[j] = 0.0f;
        float m20 = 0.0f, m21 = 0.0f;   // mem2 rows 0,1 (valid in lanes 0..15)

        const int sbase = n * kSpkStride + 8 * hi;      // spkT write base (transposed)
        float* const mem_out = out + (size_t)kSteps * kB * kDout;

        for (int it = 0; it <= kG; ++it) {
            if (it > 0) {
                const int g = it - 1;
                const float (*buf)[kElems] = tile[g & 1];
                for (int t = 0; t < kT; ++t) {
                    const int s = g * kT + t;
                    const float* sh = buf[t];

                    // ---- layer 1: c = beta*mem1 - spike(mem1-TH)*TH + b1, += x_t @ W1^T
                    v8f c;
#pragma unroll
                    for (int j = 0; j < 8; ++j) {
                        const float m = mem1[j];
                        c[j] = kBeta * m - ((m > kTh) ? kTh : 0.0f) + b1n;
                    }
#pragma unroll
                    for (int kb = 0; kb < kKB; ++kb) {
                        const int d0 = 4 * kb + 2 * hi;
                        const v2f a = *(const v2f*)(sh + n * kDin + d0);
                        c = __builtin_amdgcn_wmma_f32_16x16x4_f32(
                                false, a, false, bfrag[kb], (short)0, c, false, false);
                    }
                    mem1 = c;

                    // ---- spk1 -> LDS transposed tile spkT[h][m]
#pragma unroll
                    for (int j = 0; j < 8; ++j) {
                        spkT[sbase + j] = (mem1[j] > kTh) ? 1.0f : 0.0f;
                    }

                    // ---- layer 2 as 2x WMMA: c2 = f(mem2) + spk1 @ W2^T (transposed form)
                    v8f c2;
                    c2[0] = (hi == 0) ? (kBeta * m20 - ((m20 > kTh) ? kTh : 0.0f) + b20) : 0.0f;
                    c2[1] = (hi == 0) ? (kBeta * m21 - ((m21 > kTh) ? kTh : 0.0f) + b21) : 0.0f;
#pragma unroll
                    for (int j = 2; j < 8; ++j) c2[j] = 0.0f;
#pragma unroll
                    for (int kb = 0; kb < 2; ++kb) {
                        const int k0 = 4 * kb + 2 * hi;
                        v2f bb;
                        bb.x = spkT[(k0    ) * kSpkStride + n];
                        bb.y = spkT[(k0 + 1) * kSpkStride + n];
                        c2 = __builtin_amdgcn_wmma_f32_16x16x4_f32(
                                false, w2a[kb], false, bb, (short)0, c2, false, false);
                    }
                    m20 = c2[0];
                    m21 = c2[1];

                    // ---- record (spk2, mem2): lane n (<16) owns batch row b0+n, both outputs
                    if (hi == 0) {
                        v2f so, mo;
                        so.x = (m20 > kTh) ? 1.0f : 0.0f;
                        so.y = (m21 > kTh) ? 1.0f : 0.0f;
                        mo.x = m20;
                        mo.y = m21;
                        const int idx = (s * kB + b0 + n) * kDout;
                        *(v2f*)(out + idx)     = so;   // coalesced 128B store across lanes
                        *(v2f*)(mem_out + idx) = mo;
                    }
                }
            }
            __syncthreads();
        }
    } else {
        // ---------------- loader waves (waves 1..3, 96 threads) ----------------
        const int lt = tid - 32;  // 0..95
        int goffs[15], loffs[15];
#pragma unroll
        for (int i = 0; i < 15; ++i) {
            const int e  = lt + i * 96;          // flat tile element 0..1439
            const int r  = e / kDin;
            const int d  = e - r * kDin;
            const int l  = d / kCh;
            const int cc = d - l * kCh;
            goffs[i] = (((b0 + r) * kLvl + l) * kSteps) * kCh + cc;  // + s*kCh per step
            loffs[i] = r * kDin + d;
        }
        const bool hasR = (lt < kElems - 15 * 96);   // remainder elems 1440..1471
        int goffR, loffR;
        {
            const int e  = 15 * 96 + lt;
            const int r  = e / kDin;
            const int d  = e - r * kDin;
            const int l  = d / kCh;
            const int cc = d - l * kCh;
            goffR = (((b0 + r) * kLvl + l) * kSteps) * kCh + cc;
            loffR = r * kDin + d;
        }

        for (int it = 0; it <= kG; ++it) {
            if (it < kG) {
                for (int t = 0; t < kT; ++t) {
                    float* dst = &tile[it & 1][t][0];
                    const int soff = (it * kT + t) * kCh;
#if USE_ASYNC_LDS
#pragma unroll
                    for (int i = 0; i < 15; ++i) {
                        async_ld_f32(x + goffs[i] + soff, dst + loffs[i]);
                    }
                    if (hasR) async_ld_f32(x + goffR + soff, dst + loffR);
#else
#pragma unroll
                    for (int i = 0; i < 15; ++i) {
                        dst[loffs[i]] = x[goffs[i] + soff];
                    }
                    if (hasR) dst[loffR] = x[goffR + soff];
#endif
                }
#if USE_ASYNC_LDS
                wait_async0();
#endif
            }
            __syncthreads();
        }
    }
}

extern "C" void kernel_launch(void* const* d_in, const int* in_sizes, int n_in,
                              void* d_out, int out_size, void* d_ws, size_t ws_size,
                              hipStream_t stream) {
    (void)in_sizes; (void)n_in; (void)out_size; (void)d_ws; (void)ws_size;
    const float* x  = (const float*)d_in[0];
    const float* W1 = (const float*)d_in[1];
    const float* b1 = (const float*)d_in[2];
    const float* W2 = (const float*)d_in[3];
    const float* b2 = (const float*)d_in[4];
    float* out = (float*)d_out;

    snn_scan_kernel<<<dim3(kB / kTileM), dim3(128), 0, stream>>>(x, W1, b1, W2, b2, out);
}